// SpatialAlways_6236292514516
// MI455X (gfx1250) — compile-verified
//
#include <hip/hip_runtime.h>

// Sliding-window (W=128) soft-min:  res[b,t] = -tau * logsumexp(-x[b,t..t+127]/tau)
// MI455X plan: TDM bulk DMA (tensor_load_to_lds) -> LDS tile (128x reuse per element),
// s_wait_tensorcnt sync, native v_exp_f32 / v_log_f32, 4-way split accumulators
// so TRANS exp ops co-execute with the adds of the other streams.

#define TAUV   0.01f
#define BIGV   1000000000.0f
#define WIN    128
#define CHUNK  2048
#define NLDS   (CHUNK + WIN)   // 2176 floats = 8704 B of LDS
#define TPB    256             // 8 waves (wave32)

typedef unsigned int u32x4 __attribute__((ext_vector_type(4)));
typedef int          i32x8 __attribute__((ext_vector_type(8)));
typedef int          i32x4 __attribute__((ext_vector_type(4)));

#if defined(__has_builtin)
#if __has_builtin(__builtin_amdgcn_tensor_load_to_lds) && \
    __has_builtin(__builtin_amdgcn_s_wait_tensorcnt)
#define USE_TDM 1
#endif
#endif

__device__ __forceinline__ float fast_exp2(float x) {
#if defined(__has_builtin) && __has_builtin(__builtin_amdgcn_exp2f)
  return __builtin_amdgcn_exp2f(x);   // v_exp_f32
#else
  return exp2f(x);
#endif
}

__device__ __forceinline__ float fast_log2(float x) {
#if defined(__has_builtin) && __has_builtin(__builtin_amdgcn_logf)
  return __builtin_amdgcn_logf(x);    // v_log_f32
#else
  return log2f(x);
#endif
}

__global__ __launch_bounds__(TPB) void softmin_window_kernel(
    const float* __restrict__ in, float* __restrict__ out, int T)
{
  __shared__ float sw[NLDS];

  const int b        = blockIdx.y;
  const int rowStart = blockIdx.x * CHUNK;
  const float* rowPtr = in + (size_t)b * (size_t)T + rowStart;

  // #elements of this tile still inside the row (rest is OOB -> padded BIG)
  const int remI = (T - rowStart < NLDS) ? (T - rowStart) : NLDS;
  const unsigned rem = (unsigned)remI;

#ifdef USE_TDM
  if (threadIdx.x < 32u) {  // one wave issues the TDM descriptor
    unsigned long long ga = (unsigned long long)(uintptr_t)rowPtr;
    unsigned ldsAddr = (unsigned)(uintptr_t)&sw[0];  // addr[31:0] = LDS offset

    // D# group 0: count=1 | lds_addr | global_addr[56:0] | type=2
    u32x4 g0;
    g0.x = 1u;
    g0.y = ldsAddr;
    g0.z = (unsigned)(ga & 0xFFFFFFFFull);
    g0.w = ((unsigned)(ga >> 32) & 0x01FFFFFFu) | 0x80000000u; // type=2 in [127:126]

    // D# group 1: wg_mask=0, data_size=2 (4B), tensor_dim0=rem (OOB reads -> 0),
    // tensor_dim1=1, tile_dim0=NLDS, tile_dim1/2=0, dim0_stride=NLDS.
    i32x8 g1;
    g1[0] = (int)0x00020000u;                            // data_size = 4B
    g1[1] = (int)((rem & 0xFFFFu) << 16);                // tensor_dim0[15:0]
    g1[2] = (int)(((rem >> 16) & 0xFFFFu) | (1u << 16)); // tensor_dim0[31:16] | tensor_dim1=1
    g1[3] = (int)((unsigned)NLDS << 16);                 // tile_dim0 = 2176
    g1[4] = 0;                                           // tile_dim1=0, tile_dim2=0
    g1[5] = (int)(unsigned)NLDS;                         // tensor_dim0_stride[31:0]
    g1[6] = 0;
    g1[7] = 0;

    i32x4 z4 = {0, 0, 0, 0};
#if __clang_major__ >= 23
    i32x8 z8 = {0, 0, 0, 0, 0, 0, 0, 0};
    __builtin_amdgcn_tensor_load_to_lds(g0, g1, z4, z4, z8, 0);
#else
    __builtin_amdgcn_tensor_load_to_lds(g0, g1, z4, z4, 0);
#endif
    __builtin_amdgcn_s_wait_tensorcnt(0);                // TENSORcnt == 0
  }
  __syncthreads();  // TDM data now visible to all 8 waves

  // Patch OOB-zero tail to BIG so padding contributes exp(-inf) = 0.
  for (int i = (int)threadIdx.x; i < NLDS; i += TPB)
    if (i >= remI) sw[i] = BIGV;
  __syncthreads();
#else
  for (int i = (int)threadIdx.x; i < NLDS; i += TPB)
    sw[i] = (i < remI) ? rowPtr[i] : BIGV;
  __syncthreads();
#endif

  // res = m - tau*ln2 * log2( sum_j exp2( (m - x_j) * log2(e)/tau ) )
  const float k = 144.26950408889634f;        // log2(e) / tau
  const float c = 0.0069314718055994531f;     // tau * ln(2)
  float* o = out + (size_t)b * (size_t)T + rowStart;

  for (int r = 0; r < CHUNK / TPB; ++r) {
    const int li = r * TPB + (int)threadIdx.x;
    const float* w = &sw[li];

    // ---- pass 1: window min, 4 independent chains (breaks v_min dep chain) ----
    float m0 = w[0], m1 = w[1], m2 = w[2], m3 = w[3];
    #pragma unroll
    for (int j = 4; j < WIN; j += 4) {
      m0 = fminf(m0, w[j + 0]);
      m1 = fminf(m1, w[j + 1]);
      m2 = fminf(m2, w[j + 2]);
      m3 = fminf(m3, w[j + 3]);
    }
    const float m = fminf(fminf(m0, m1), fminf(m2, m3));

    // ---- pass 2: exp2 sum, 4 independent accumulators (TRANS/VALU overlap) ----
    float s0 = 0.0f, s1 = 0.0f, s2 = 0.0f, s3 = 0.0f;
    #pragma unroll
    for (int j = 0; j < WIN; j += 4) {
      s0 += fast_exp2((m - w[j + 0]) * k);
      s1 += fast_exp2((m - w[j + 1]) * k);
      s2 += fast_exp2((m - w[j + 2]) * k);
      s3 += fast_exp2((m - w[j + 3]) * k);
    }
    const float s = (s0 + s1) + (s2 + s3);

    o[li] = m - c * fast_log2(s);   // s in [1,128], log never sees 0
  }
}

extern "C" void kernel_launch(void* const* d_in, const int* in_sizes, int n_in,
                              void* d_out, int out_size, void* d_ws, size_t ws_size,
                              hipStream_t stream) {
  const float* in = (const float*)d_in[0];
  float* out = (float*)d_out;
  const int T = 65536;
  const int B = in_sizes[0] / T;          // 16
  dim3 grid(T / CHUNK, B);                // 32 x 16 = 512 blocks
  softmin_window_kernel<<<grid, TPB, 0, stream>>>(in, out, T);
}